// FastSlidingCMVN_64647847740094
// MI455X (gfx1250) — compile-verified
//
#include <hip/hip_runtime.h>
#include <hip/hip_bf16.h>

typedef __attribute__((ext_vector_type(2))) float f2;

// feats [B=64, T=8192, C=80] f32; sliding CMVN, window K=301, reflect pad.
constexpr int Bb   = 64;
constexpr int Tt   = 8192;
constexpr int Cc   = 80;
constexpr int CG   = Cc / 2;      // 40 float2 channel-groups
constexpr int PAD  = 150;         // K/2
constexpr int LCH  = 512;         // t-chunk per thread
constexpr int NCH  = Tt / LCH;    // 16
constexpr int BLK  = 128;         // threads per block (4 waves)
constexpr int RING = 32;          // ring depth (f2 slots per thread)
constexpr int HALF = 16;          // double-buffer half
constexpr float INVK = 1.0f / 301.0f;
constexpr float EPSV = 1e-8f;

__device__ __forceinline__ int iabs(int a) { return a < 0 ? -a : a; }

__device__ __forceinline__ int refl_init(int j) {  // full 2-sided reflect (init only)
  int a = iabs(j);
  return a >= Tt ? 2 * (Tt - 1) - a : a;
}

__global__ __launch_bounds__(BLK) void cmvn_v2_kernel(
    const float* __restrict__ in, float* __restrict__ out) {
  // Per-thread 32-deep float2 enter-stream ring; layout ring[slot*BLK+tid]
  // => b64 LDS reads/writes hit all 64 banks with zero conflicts.
  __shared__ f2 ring[RING * BLK];

  const int tid   = threadIdx.x;
  const int g     = blockIdx.x * BLK + tid;
  const int cg    = g % CG;                  // lanes -> consecutive channel pairs
  const int r     = g / CG;
  const int chunk = r % NCH;
  const int b     = r / NCH;
  const int t0    = chunk * LCH;

  const float* __restrict__ ibase = in  + (size_t)b * Tt * Cc + cg * 2;
  float*       __restrict__ obase = out + (size_t)b * Tt * Cc + cg * 2;

  // LDS byte address of this thread's ring column (low 32 bits of generic ptr
  // == offset within the workgroup LDS allocation).
  const unsigned ringBase =
      (unsigned)(size_t)(const void*)&ring[0] + (unsigned)tid * 8u;

  // ---- init window sums over [t0-PAD, t0+PAD] (reflected), b64 loads ----
  f2 S = {0.f, 0.f}, S2 = {0.f, 0.f};
#pragma unroll 8
  for (int d = -PAD; d <= PAD; ++d) {
    const f2 x = *(const f2*)(ibase + (size_t)refl_init(t0 + d) * Cc);
    S += x;
    S2 = __builtin_elementwise_fma(x, x, S2);
  }

  // Stage one half (HALF steps) of the enter stream asynchronously into LDS.
  // Enter index for step t is t+PAD+1 in [151, 8342]: only upper reflect.
  auto issue_half = [&](int slotBase, int tFirst) {
    unsigned la = ringBase + (unsigned)slotBase * (BLK * 8u);
#pragma unroll
    for (int j = 0; j < HALF; ++j) {
      const int e = (Tt - 1) - iabs(tFirst + j + PAD + 1 - (Tt - 1));
      const float* ga = ibase + (size_t)e * Cc;
      asm volatile("global_load_async_to_lds_b64 %0, %1, off"
                   :: "v"(la), "v"(ga)
                   : "memory");
      la += BLK * 8u;
    }
  };

  issue_half(0, t0);
  issue_half(HALF, t0 + HALF);

  constexpr int nBlk = LCH / HALF;  // 32
  for (int blk = 0; blk < nBlk; ++blk) {
    // Async loads complete in order: outstanding<=2*HALF, waiting <=HALF
    // guarantees the half we are about to read has landed. Last block drains.
    if (blk < nBlk - 1)
      asm volatile("s_wait_asynccnt 0x10" ::: "memory");
    else
      asm volatile("s_wait_asynccnt 0x0" ::: "memory");

    const int slotBase = (blk & 1) * HALF;
#pragma unroll
    for (int j = 0; j < HALF; ++j) {
      const int t  = t0 + blk * HALF + j;
      const f2 xe  = ring[(slotBase + j) * BLK + tid];          // enter (LDS)
      const f2 x   = *(const f2*)(ibase + (size_t)t * Cc);      // center (cache hit)
      const int lv = iabs(t - PAD);                             // low-edge reflect only
      const f2 xl  = *(const f2*)(ibase + (size_t)lv * Cc);     // leave (L2 hit)

      const f2 mu = S * INVK;
      f2 var = __builtin_elementwise_fma(-mu, mu, S2 * INVK);   // E[x^2]-mu^2
      var = __builtin_elementwise_max(var, (f2){EPSV, EPSV});
      f2 inv; inv.x = rsqrtf(var.x); inv.y = rsqrtf(var.y);     // TRANS, co-executes
      const f2 y = (x - mu) * inv;
      __builtin_nontemporal_store(y, (f2*)(obase + (size_t)t * Cc));

      S  += xe - xl;
      S2 += __builtin_elementwise_fma(xe, xe, -(xl * xl));
    }
    if (blk < nBlk - 2) issue_half(slotBase, t0 + (blk + 2) * HALF);
  }
}

extern "C" void kernel_launch(void* const* d_in, const int* in_sizes, int n_in,
                              void* d_out, int out_size, void* d_ws, size_t ws_size,
                              hipStream_t stream) {
  (void)in_sizes; (void)n_in; (void)out_size; (void)d_ws; (void)ws_size;
  const float* feats = (const float*)d_in[0];
  float* out = (float*)d_out;
  const int threads = Bb * CG * NCH;           // 40960
  dim3 grid(threads / BLK), block(BLK);        // 320 x 128
  cmvn_v2_kernel<<<grid, block, 0, stream>>>(feats, out);
}